// FSAIntegratedInputLayer_16862041604529
// MI455X (gfx1250) — compile-verified
//
#include <hip/hip_runtime.h>
#include <stdint.h>

// Problem constants (match reference)
#define BB 64
#define SS 4096
#define VV 1000
#define NSS 512          // number of FSA states
#define DWD 512          // word emb dim
#define DFD 256          // fsa emb dim
#define DOUT 768         // DWD + DFD
#define TS 16            // sequence positions per block in emit kernel
#define KSEG 16          // segments per chain (speculative scan)
#define LSEG (SS/KSEG)   // 256 steps per segment

typedef uint32_t u32x4 __attribute__((ext_vector_type(4)));
typedef uint32_t u32x8 __attribute__((ext_vector_type(8)));

// ---------------------------------------------------------------------------
// Fallback: sequential FSA scan (used only if workspace is too small).
// ---------------------------------------------------------------------------
__global__ __launch_bounds__(32) void fsa_scan_kernel(
    const int* __restrict__ wid,
    const int* __restrict__ trans,
    int* __restrict__ states)
{
    int b = blockIdx.x * 32 + threadIdx.x;
    if (b >= BB) return;
    const int* w = wid + (size_t)b * SS;
    int* o = states + (size_t)b * SS;
    int st = 0;
    for (int s = 0; s < SS; s += 4) {
        int4 w4 = *(const int4*)(w + s);
        st = trans[st * VV + w4.x]; o[s + 0] = st;
        st = trans[st * VV + w4.y]; o[s + 1] = st;
        st = trans[st * VV + w4.z]; o[s + 2] = st;
        st = trans[st * VV + w4.w]; o[s + 3] = st;
    }
}

// ---------------------------------------------------------------------------
// Phase 1: speculative segment maps (512 entry states per (batch, segment)).
// ---------------------------------------------------------------------------
__global__ __launch_bounds__(256) void seg_map_kernel(
    const int* __restrict__ wid,
    const int* __restrict__ trans,
    int* __restrict__ maps)
{
    __shared__ int sym[LSEG];                 // 1 KB
    const int bk = blockIdx.x;                // 0 .. B*K-1
    const int b  = bk >> 4;
    const int k  = bk & (KSEG - 1);
    const int t  = threadIdx.x;

    const int* w = wid + (size_t)b * SS + (size_t)k * LSEG;
    sym[t] = w[t];                            // LSEG == blockDim == 256
    __syncthreads();

    int s0 = t;                               // entry state chain #1
    int s1 = t + 256;                         // entry state chain #2
    for (int i = 0; i < LSEG; ++i) {
        int sy = sym[i];                      // LDS broadcast
        s0 = trans[s0 * VV + sy];
        s1 = trans[s1 * VV + sy];
    }
    maps[((size_t)bk << 9) + t]       = s0;
    maps[((size_t)bk << 9) + t + 256] = s1;
}

// ---------------------------------------------------------------------------
// Phase 2: serial composition of the 16 segment maps per batch (tiny).
// ---------------------------------------------------------------------------
__global__ __launch_bounds__(64) void seg_link_kernel(
    const int* __restrict__ maps,
    int* __restrict__ entries)
{
    int b = threadIdx.x;
    if (b >= BB) return;
    int st = 0;                               // START_STATE
    for (int k = 0; k < KSEG; ++k) {
        entries[b * KSEG + k] = st;
        st = maps[((size_t)(b * KSEG + k) << 9) + st];
    }
}

// ---------------------------------------------------------------------------
// Phase 3: re-walk each segment from its known entry state, emitting states.
// ---------------------------------------------------------------------------
__global__ __launch_bounds__(64) void seg_emit_kernel(
    const int* __restrict__ wid,
    const int* __restrict__ trans,
    const int* __restrict__ entries,
    int* __restrict__ states)
{
    int c = blockIdx.x * 64 + threadIdx.x;    // 0 .. B*K-1
    if (c >= BB * KSEG) return;
    int b = c >> 4;
    int k = c & (KSEG - 1);
    const int* w = wid + (size_t)b * SS + (size_t)k * LSEG;
    int* o = states + (size_t)b * SS + (size_t)k * LSEG;
    int st = entries[c];
    for (int i = 0; i < LSEG; i += 4) {
        int4 w4 = *(const int4*)(w + i);
        st = trans[st * VV + w4.x]; o[i + 0] = st;
        st = trans[st * VV + w4.y]; o[i + 1] = st;
        st = trans[st * VV + w4.z]; o[i + 2] = st;
        st = trans[st * VV + w4.w]; o[i + 3] = st;
    }
}

// ---------------------------------------------------------------------------
// Emit concat([we*sqrt(DW)+PE, se]) -> [B, S, 768] f32.
// Block = 256 threads, owns TS=16 sequence positions for all 64 batches.
//  - PE tile (16x512 f32, 32KB LDS) computed once per block, reused 64x.
//  - we half: float4 gather + fma + coalesced b128 store (VALU path).
//  - se half: TDM gather-mode tensor_load_to_lds pulls the 16 fsa_emb rows
//    (selected by the tile's 16 uniform state ids) into LDS in one DMA op,
//    then global_store_async_from_lds_b128 scatters them to the output.
//    TENSORcnt + ASYNCcnt + block barriers provide the ordering.
// ---------------------------------------------------------------------------
__global__ __launch_bounds__(256) void emit_kernel(
    const int* __restrict__ wid,
    const float* __restrict__ word_emb,
    const float* __restrict__ fsa_emb,
    const int* __restrict__ states,
    float* __restrict__ out)
{
    __shared__ float  pe[TS * DWD];     // 32 KB
    __shared__ float4 sebuf[1024];      // 16 KB TDM landing zone: [16 rows][64 float4]

    const int t  = threadIdx.x;
    const int s0 = blockIdx.x * TS;

    // ---- positional-encoding tile into LDS -------------------------------
    const float kNegLogOverD = -9.210340371976184f / (float)DWD;  // -ln(1e4)/512
    for (int k = 0; k < (TS * DWD) / 256; ++k) {
        int idx = t + k * 256;
        int sl  = idx >> 9;            // /512
        int d   = idx & (DWD - 1);
        int h2  = d & ~1;              // 2h
        float freq = expf(kNegLogOverD * (float)h2);
        float ang  = (float)(s0 + sl) * freq;
        pe[idx] = (d & 1) ? cosf(ang) : sinf(ang);
    }
    __syncthreads();

    const float scale = 22.62741699796952f;               // sqrt(512)
    const float4* we4 = (const float4*)word_emb;          // 128 float4 per row
    const float4* pe4 = (const float4*)pe;
    float4* out4 = (float4*)out;                          // 192 float4 per row

    // wave-relative LDS byte offsets (flat-pointer low 32 bits == LDS offset)
    const uint32_t lds_base = (uint32_t)(uintptr_t)(&sebuf[0]);
    const uint32_t mybase   = (uint32_t)(uintptr_t)(&sebuf[t]);

    // ---- constant part of the TDM gather descriptor ----------------------
    // D# group0: count=1 | gather_mode=1 | 16-bit indices; lds_addr; fsa_emb base
    const uint64_t ga = (uint64_t)(uintptr_t)fsa_emb;
    const u32x4 g0 = { 0x80000001u,                       // count=1, gather_mode=1
                       lds_base,
                       (uint32_t)ga,
                       (((uint32_t)(ga >> 32)) & 0x01FFFFFFu) | 0x80000000u }; // type=2
    // D# group1: data_size=4B; tensor_dim0=256; tensor_dim1=512;
    //            tile_dim0=256; tile_dim1=16 indices; tensor_dim0_stride=256
    const u32x8 g1 = { 0x00020000u,
                       (uint32_t)(DFD) << 16,             // tensor_dim0 lo16
                       (uint32_t)(NSS) << 16,             // tensor_dim1 lo16
                       (uint32_t)(DFD) << 16,             // tile_dim0
                       (uint32_t)TS,                      // tile_dim1 = #indices
                       (uint32_t)DFD,                     // tensor_dim0_stride lo32
                       0u, 0u };

    for (int b = 0; b < BB; ++b) {
        const size_t rowbase = (size_t)b * SS + s0;

        // each wave: its own async stores from the previous iteration must be
        // done reading sebuf; then block-wide barrier so TDM may overwrite it
        asm volatile("s_wait_asynccnt 0" ::: "memory");
        __syncthreads();

        // ---- wave 0 issues the TDM gather of 16 fsa_emb rows -------------
        if (t < 32) {
            uint32_t p[8];
#pragma unroll
            for (int i = 0; i < 8; ++i) {
                uint32_t lo = (uint32_t)__builtin_amdgcn_readfirstlane(
                                  states[rowbase + 2 * i]);
                uint32_t hi = (uint32_t)__builtin_amdgcn_readfirstlane(
                                  states[rowbase + 2 * i + 1]);
                p[i] = (lo & 0xFFFFu) | (hi << 16);
            }
            u32x4 g2 = { p[0], p[1], p[2], p[3] };        // row indices 0..7
            u32x4 g3 = { p[4], p[5], p[6], p[7] };        // row indices 8..15
            asm volatile("tensor_load_to_lds %0, %1, %2, %3"
                         :: "s"(g0), "s"(g1), "s"(g2), "s"(g3) : "memory");
        }

        // ---- we half (overlaps with the TDM gather) ----------------------
#pragma unroll
        for (int j = 0; j < 8; ++j) {
            int e  = t + j * 256;          // 0..2047
            int sl = e >> 7;               // /128 -> tile row
            int c4 = e & 127;              // float4 column
            int w  = wid[rowbase + sl];
            float4 v = we4[(size_t)w * 128 + c4];
            float4 p = pe4[sl * 128 + c4];
            float4 r;
            r.x = fmaf(v.x, scale, p.x);
            r.y = fmaf(v.y, scale, p.y);
            r.z = fmaf(v.z, scale, p.z);
            r.w = fmaf(v.w, scale, p.w);
            out4[(rowbase + sl) * 192 + c4] = r;
        }

        // ---- TDM done -> rows are in LDS; scatter them to the output -----
        if (t < 32) {
            __builtin_amdgcn_s_wait_tensorcnt(0);
        }
        __syncthreads();
#pragma unroll
        for (int j = 0; j < 4; ++j) {
            int e  = t + j * 256;          // 0..1023 = sl*64 + c4
            int sl = e >> 6;
            int c4 = e & 63;
            float4* gdst = out4 + (rowbase + sl) * 192 + 128 + c4;
            uint32_t loff = mybase + (uint32_t)j * 4096;  // slot (t + j*256)
            asm volatile("global_store_async_from_lds_b128 %0, %1, off"
                         :: "v"(gdst), "v"(loff) : "memory");
        }
    }
    asm volatile("s_wait_asynccnt 0" ::: "memory");
}

// ---------------------------------------------------------------------------
// inputs (setup_inputs order):
//   d_in[0] word_id_sequence [B,S]  int32
//   d_in[1] word_emb         [V,DW] f32
//   d_in[2] fsa_emb          [NS,DF] f32
//   d_in[3] fsa_transitions  [NS,V] int32
// d_out: [B,S,768] f32.
// d_ws layout: states [B,S] i32 (1 MB) | maps [B,K,512] i32 (2 MB) |
//              entries [B,K] i32 (4 KB)
// ---------------------------------------------------------------------------
extern "C" void kernel_launch(void* const* d_in, const int* in_sizes, int n_in,
                              void* d_out, int out_size, void* d_ws, size_t ws_size,
                              hipStream_t stream)
{
    const int*   wid      = (const int*)d_in[0];
    const float* word_emb = (const float*)d_in[1];
    const float* fsa_emb  = (const float*)d_in[2];
    const int*   trans    = (const int*)d_in[3];
    float*       out      = (float*)d_out;

    int* states = (int*)d_ws;
    const size_t states_bytes  = (size_t)BB * SS * 4;            // 1 MB
    const size_t maps_bytes    = (size_t)BB * KSEG * NSS * 4;    // 2 MB
    const size_t entries_bytes = (size_t)BB * KSEG * 4;          // 4 KB

    if (ws_size >= states_bytes + maps_bytes + entries_bytes) {
        int* maps    = (int*)((char*)d_ws + states_bytes);
        int* entries = (int*)((char*)maps + maps_bytes);
        seg_map_kernel <<<BB * KSEG, 256, 0, stream>>>(wid, trans, maps);
        seg_link_kernel<<<1, 64, 0, stream>>>(maps, entries);
        seg_emit_kernel<<<(BB * KSEG) / 64, 64, 0, stream>>>(wid, trans, entries, states);
    } else {
        fsa_scan_kernel<<<2, 32, 0, stream>>>(wid, trans, states);
    }

    emit_kernel<<<SS / TS, 256, 0, stream>>>(wid, word_emb, fsa_emb, states, out);
}